// FeatureDecorr_37855841747395
// MI455X (gfx1250) — compile-verified
//
#include <hip/hip_runtime.h>

// ---------------------------------------------------------------------------
// FeatureDecorr (group whitening) for MI455X / gfx1250, wave32.
//
//   k_gram : one pass over x -> group sums S[n][64] + Gram P[n][64][64]
//            (WMMA f32 16x16x4, double-buffered GLOBAL_LOAD_ASYNC_TO_LDS)
//   k_ns   : per n: cov = P/M - mean meanT + eps I, Newton-Schulz x10 (WMMA),
//            W' = weight1 @ (Z / sqrt(||cov||_F)),  c = bias1 - W' @ mean
//   k_apply: y = W' @ x1 + c (WMMA, async double-buffered loads, LDS round
//            trip for coalesced float4 stores)
// ---------------------------------------------------------------------------

typedef __attribute__((ext_vector_type(2))) float v2f;
typedef __attribute__((ext_vector_type(8))) float v8f;

#define NB    32          // batch
#define GD    64          // groups
#define HWSZ  2304        // 48*48
#define COUT  8           // 512 / 64
#define MTOT  18432       // COUT * HWSZ
#define LDP   68          // LDS pitch (floats): 16B-aligned rows, no bank
                          // conflicts on K-column fragment reads
#define EPSV  1e-5f
#define NITER 10

// workspace layout in floats
#define WS_S  0                        // [32][64]   group sums (atomic)
#define WS_P  (NB*GD)                  // [32][4096] gram       (atomic)
#define WS_W  (WS_P + NB*GD*GD)        // [32][4096] W' = weight1 @ decorr
#define WS_C  (WS_W + NB*GD*GD)        // [32][64]   c  = bias1 - W' @ mean
#define WS_ZERO_BYTES ((size_t)WS_W * sizeof(float))   // zero S and P only

// --- CDNA5 async global->LDS helpers ---------------------------------------
// lds_off: LDS byte offset (low 32 bits of the generic shared-space address).
__device__ __forceinline__ void async_load_b128(unsigned lds_off,
                                                const void* gaddr) {
  unsigned long long a = (unsigned long long)gaddr;
  asm volatile("global_load_async_to_lds_b128 %0, %1, off"
               :: "v"(lds_off), "v"(a)
               : "memory");
}
__device__ __forceinline__ void wait_async0() {
  asm volatile("s_wait_asynccnt 0x0" ::: "memory");
}
__device__ __forceinline__ unsigned lds_off32(const void* shared_ptr) {
  return (unsigned)(unsigned long long)shared_ptr;   // addr[31:0] = LDS offset
}

// --- WMMA fragment helpers (fp32, 16x16x4) ---------------------------------
// A 16x4 (MxK): lanes 0-15 row M=l (K, K+1 in .x/.y); lanes 16-31 K+2, K+3.
__device__ __forceinline__ v2f frag_colk(const float* Ms, int r0, int k,
                                         int l16, int koff) {
  v2f a;
  a.x = Ms[(r0 + l16) * LDP + k + koff];
  a.y = Ms[(r0 + l16) * LDP + k + 1 + koff];
  return a;
}
// B 4x16 (KxN): lanes 0-15 col N=l (rows K, K+1); lanes 16-31 rows K+2, K+3.
__device__ __forceinline__ v2f frag_rowk(const float* Ms, int c0, int k,
                                         int l16, int koff) {
  v2f b;
  b.x = Ms[(k + koff) * LDP + c0 + l16];
  b.y = Ms[(k + 1 + koff) * LDP + c0 + l16];
  return b;
}

__device__ __forceinline__ v8f wmma_f32(v2f a, v2f b, v8f c) {
  return __builtin_amdgcn_wmma_f32_16x16x4_f32(false, a, false, b,
                                               (short)0, c, false, false);
}

// 64x64 LDS matmul D = A @ B, 16 waves (512 threads), one 16x16 tile/wave.
__device__ __forceinline__ void mm64(float* D, const float* A, const float* B) {
  const int tid  = threadIdx.x;
  const int lane = tid & 31;
  const int w    = tid >> 5;
  const int l16  = lane & 15;
  const int koff = (lane >> 4) * 2;
  const int gi   = (w >> 2) * 16;
  const int gj   = (w & 3) * 16;
  __syncthreads();
  v8f acc = {};
#pragma unroll
  for (int k = 0; k < 64; k += 4) {
    v2f a = frag_colk(A, gi, k, l16, koff);
    v2f b = frag_rowk(B, gj, k, l16, koff);
    acc = wmma_f32(a, b, acc);
  }
  __syncthreads();
#pragma unroll
  for (int r = 0; r < 8; ++r) {
    int row = gi + r + (lane >> 4) * 8;   // C/D layout: M = r (+8 upper half)
    D[row * LDP + gj + l16] = acc[r];
  }
  __syncthreads();
}

// ---------------------------------------------------------------------------
// Kernel 1: group sums + Gram. grid = 32*8*2 (n, c_outer, half), block = 512.
// Double-buffered async global->LDS staging; 18 tiles of 64 hw per block.
// ---------------------------------------------------------------------------
__global__ __launch_bounds__(512) void k_gram(const float* __restrict__ x,
                                              float* __restrict__ ws) {
  __shared__ __align__(16) float Xs[2][GD * LDP];
  __shared__ float rsum[512];

  const int bid  = blockIdx.x;
  const int n    = bid >> 4;
  const int sb   = bid & 15;
  const int co   = sb >> 1;
  const int half = sb & 1;
  const int tid  = threadIdx.x;

  const int g   = tid >> 3;     // 0..63 group row
  const int sub = tid & 7;      // 8 threads per 64-float row segment
  const float* base = x + (size_t)(n * 512 + co * GD + g) * HWSZ
                        + half * (HWSZ / 2) + sub * 8;

  const unsigned ldsRow0 = lds_off32(&Xs[0][g * LDP + sub * 8]);
  const unsigned bufB    = (unsigned)(GD * LDP * sizeof(float));

  const int lane = tid & 31;
  const int w    = tid >> 5;
  const int l16  = lane & 15;
  const int koff = (lane >> 4) * 2;
  const int gi   = (w >> 2) * 16;
  const int gj   = (w & 3) * 16;

  const int NT = HWSZ / (2 * 64);   // 18 tiles

  // prologue: async-fill tile 0
  async_load_b128(ldsRow0, base);
  async_load_b128(ldsRow0 + 16, base + 4);
  wait_async0();
  __syncthreads();

  float psum = 0.f;
  v8f acc = {};

  for (int t = 0; t < NT; ++t) {
    if (t + 1 < NT) {                       // prefetch next tile (other buf)
      unsigned dst = ldsRow0 + ((t + 1) & 1) * bufB;
      const float* src = base + (t + 1) * 64;
      async_load_b128(dst, src);
      async_load_b128(dst + 16, src + 4);
    }
    const float* Xc = Xs[t & 1];
    const float* myrow = &Xc[g * LDP + sub * 8];
#pragma unroll
    for (int i = 0; i < 8; ++i) psum += myrow[i];   // group sums
#pragma unroll
    for (int k = 0; k < 64; k += 4) {
      v2f a = frag_colk(Xc, gi, k, l16, koff);
      v2f b = frag_colk(Xc, gj, k, l16, koff);  // Gram: B[k][q] = X[gj+q][k]
      acc = wmma_f32(a, b, acc);
    }
    wait_async0();
    __syncthreads();
  }

  float* P = ws + WS_P + (size_t)n * GD * GD;
#pragma unroll
  for (int r = 0; r < 8; ++r) {
    int row = gi + r + (lane >> 4) * 8;
    atomicAdd(&P[row * GD + gj + l16], acc[r]);
  }

  rsum[tid] = psum;
  __syncthreads();
  if (tid < GD) {
    float s = 0.f;
#pragma unroll
    for (int i = 0; i < 8; ++i) s += rsum[tid * 8 + i];
    atomicAdd(ws + WS_S + n * GD + tid, s);
  }
}

// ---------------------------------------------------------------------------
// Kernel 2: Newton-Schulz inverse sqrt + fold affine. grid = 32, block = 512.
// ---------------------------------------------------------------------------
__global__ __launch_bounds__(512) void k_ns(const float* __restrict__ weight1,
                                            const float* __restrict__ bias1,
                                            float* __restrict__ ws) {
  __shared__ __align__(16) float B0[GD * LDP];
  __shared__ __align__(16) float B1[GD * LDP];
  __shared__ __align__(16) float B2[GD * LDP];
  __shared__ __align__(16) float B3[GD * LDP];
  __shared__ float meanS[GD];
  __shared__ float snorm;

  const int n   = blockIdx.x;
  const int tid = threadIdx.x;
  const float invM = 1.0f / (float)MTOT;

  float* Yp = B0;
  float* Zp = B1;
  float* Tp = B2;
  float* Up = B3;
  float* red = B3;   // B3 unused until first mm64 -> reuse as reduction buf

  if (tid < GD) meanS[tid] = ws[WS_S + n * GD + tid] * invM;
  __syncthreads();

  // cov into Yp, Frobenius norm^2 partial
  float ss = 0.f;
#pragma unroll
  for (int i = 0; i < 8; ++i) {
    int e = tid * 8 + i;
    int r = e >> 6, c = e & 63;
    float cv = ws[WS_P + (size_t)n * GD * GD + e] * invM
               - meanS[r] * meanS[c] + ((r == c) ? EPSV : 0.f);
    Yp[r * LDP + c] = cv;
    ss += cv * cv;
  }
  red[tid] = ss;
  __syncthreads();
  for (int s = 256; s > 0; s >>= 1) {
    if (tid < s) red[tid] += red[tid + s];
    __syncthreads();
  }
  if (tid == 0) snorm = sqrtf(red[0]);
  __syncthreads();
  const float nrm = snorm;
  const float innrm = 1.0f / nrm;

  // Y = cov / ||cov||_F ;  Z = I
#pragma unroll
  for (int i = 0; i < 8; ++i) {
    int e = tid * 8 + i;
    int r = e >> 6, c = e & 63;
    Yp[r * LDP + c] *= innrm;
    Zp[r * LDP + c] = (r == c) ? 1.f : 0.f;
  }
  __syncthreads();

  for (int it = 0; it < NITER; ++it) {
    mm64(Up, Zp, Yp);                      // U = Z @ Y
#pragma unroll
    for (int i = 0; i < 8; ++i) {          // T = 1.5 I - 0.5 U
      int e = tid * 8 + i;
      int r = e >> 6, c = e & 63;
      Tp[r * LDP + c] = ((r == c) ? 1.5f : 0.f) - 0.5f * Up[r * LDP + c];
    }
    __syncthreads();
    mm64(Up, Yp, Tp);                      // newY = Y @ T  (in Up)
    mm64(Yp, Tp, Zp);                      // newZ = T @ Z  (in old Y buffer)
    float* tmp = Zp; Zp = Yp; Yp = Up; Up = tmp;   // rotate
  }

  // Up <- weight1 ; Tp <- weight1 @ Z ; scale by 1/sqrt(nrm) -> W'
#pragma unroll
  for (int i = 0; i < 8; ++i) {
    int e = tid * 8 + i;
    Up[(e >> 6) * LDP + (e & 63)] = weight1[e];
  }
  mm64(Tp, Up, Zp);
  const float sc = rsqrtf(nrm);
#pragma unroll
  for (int i = 0; i < 8; ++i) {
    int e = tid * 8 + i;
    int r = e >> 6, c = e & 63;
    float wv = Tp[r * LDP + c] * sc;
    Tp[r * LDP + c] = wv;
    ws[WS_W + (size_t)n * GD * GD + e] = wv;
  }
  __syncthreads();
  if (tid < GD) {                          // c = bias1 - W' @ mean
    float acc = bias1[tid];
    for (int j = 0; j < GD; ++j) acc -= Tp[tid * LDP + j] * meanS[j];
    ws[WS_C + n * GD + tid] = acc;
  }
}

// ---------------------------------------------------------------------------
// Kernel 3: y = W' @ x1 + c. grid = 32*8*4 (n, c_outer, quarter), block=512.
// W' staged once per block; 9 tiles, async double-buffered input staging.
// ---------------------------------------------------------------------------
__global__ __launch_bounds__(512) void k_apply(const float* __restrict__ x,
                                               const float* __restrict__ ws,
                                               float* __restrict__ out) {
  __shared__ __align__(16) float Wls[GD * LDP];
  __shared__ __align__(16) float Xs[2][GD * LDP];
  __shared__ __align__(16) float Ys[GD * LDP];
  __shared__ float cs[GD];

  const int bid = blockIdx.x;
  const int n   = bid >> 5;
  const int rem = bid & 31;
  const int co  = rem >> 2;
  const int q   = rem & 3;
  const int tid = threadIdx.x;

  const int g   = tid >> 3;
  const int sub = tid & 7;
  const size_t goff = (size_t)(n * 512 + co * GD + g) * HWSZ
                      + q * (9 * 64) + sub * 8;
  const float* xb = x + goff;
  float* ob = out + goff;

  const unsigned ldsX0 = lds_off32(&Xs[0][g * LDP + sub * 8]);
  const unsigned bufB  = (unsigned)(GD * LDP * sizeof(float));

  // prologue: async-fill tile 0; stage W' and c meanwhile
  async_load_b128(ldsX0, xb);
  async_load_b128(ldsX0 + 16, xb + 4);
  const float* Wg = ws + WS_W + (size_t)n * GD * GD;
#pragma unroll
  for (int i = 0; i < 8; ++i) {
    int e = tid * 8 + i;
    Wls[(e >> 6) * LDP + (e & 63)] = Wg[e];
  }
  if (tid < GD) cs[tid] = ws[WS_C + n * GD + tid];
  wait_async0();
  __syncthreads();

  const int lane = tid & 31;
  const int w    = tid >> 5;
  const int l16  = lane & 15;
  const int koff = (lane >> 4) * 2;
  const int gi   = (w >> 2) * 16;
  const int mj   = (w & 3) * 16;

  for (int t = 0; t < 9; ++t) {
    if (t + 1 < 9) {                       // prefetch next tile (other buf)
      unsigned dst = ldsX0 + ((t + 1) & 1) * bufB;
      const float* src = xb + (t + 1) * 64;
      async_load_b128(dst, src);
      async_load_b128(dst + 16, src + 4);
    }
    const float* Xc = Xs[t & 1];
    v8f acc = {};
#pragma unroll
    for (int k = 0; k < 64; k += 4) {
      v2f a = frag_colk(Wls, gi, k, l16, koff);
      v2f b = frag_rowk(Xc, mj, k, l16, koff);
      acc = wmma_f32(a, b, acc);
    }
#pragma unroll
    for (int r = 0; r < 8; ++r) {
      int row = gi + r + (lane >> 4) * 8;
      Ys[row * LDP + mj + l16] = acc[r] + cs[row];
    }
    __syncthreads();
    {
      float* rowp = &Ys[g * LDP + sub * 8];
      float4 q0 = make_float4(rowp[0], rowp[1], rowp[2], rowp[3]);
      float4 q1 = make_float4(rowp[4], rowp[5], rowp[6], rowp[7]);
      float4* op = reinterpret_cast<float4*>(ob + t * 64);
      op[0] = q0;
      op[1] = q1;
    }
    wait_async0();
    __syncthreads();
  }
}

// ---------------------------------------------------------------------------
extern "C" void kernel_launch(void* const* d_in, const int* in_sizes, int n_in,
                              void* d_out, int out_size, void* d_ws,
                              size_t ws_size, hipStream_t stream) {
  const float* x  = (const float*)d_in[0];   // [32,512,48,48]
  const float* w1 = (const float*)d_in[1];   // [1,64,64]
  const float* b1 = (const float*)d_in[2];   // [1,64,1]
  float* out = (float*)d_out;
  float* ws  = (float*)d_ws;

  // zero the atomic accumulators (S and P regions)
  hipMemsetAsync(d_ws, 0, WS_ZERO_BYTES, stream);

  k_gram<<<NB * COUT * 2, 512, 0, stream>>>(x, ws);
  k_ns<<<NB, 512, 0, stream>>>(w1, b1, ws);
  k_apply<<<NB * COUT * 4, 512, 0, stream>>>(x, ws, out);
}